// SGDOptimizer_3427383902675
// MI455X (gfx1250) — compile-verified
//
#include <hip/hip_runtime.h>
#include <stdint.h>

// ---------------------------------------------------------------------------
// Sparse SGD w/ "smart momentum" for MI455X (gfx1250, wave32).
// Pass 1a: param + momentum stream copies via async-LDS DMA
//          (GLOBAL_LOAD_ASYNC_TO_LDS_B128 -> s_wait_asynccnt ->
//           GLOBAL_STORE_ASYNC_FROM_LDS_B128 with TH=NT on stores)
// Pass 1b: last_update int->float convert (b128 loads + NT b128 stores)
// Pass 2 : gather/update/scatter at NNZ unique indices (stream-ordered)
//
// Roofline: ~1.2 GB streamed + ~0.5-1 GB random scatter traffic at 23.3 TB/s
// => ~100 us floor; zero dense math (WMMA inapplicable). NT hints keep the
// 192 MB L2 free of write-once output lines so the scatter gathers hit L2.
// ---------------------------------------------------------------------------

typedef float v4f __attribute__((ext_vector_type(4)));
typedef int   v4i __attribute__((ext_vector_type(4)));

// Address-space-qualified vector pointers for the async-LDS builtins
// (AS1 = global, AS3 = LDS; signature probed via round-1 diagnostics).
typedef __attribute__((address_space(1))) v4i g_v4i;
typedef __attribute__((address_space(3))) v4i l_v4i;

#define BLOCK 256

#define K_LR   0.01f
#define K_WD   1e-4f
#define K_LOG2_MOM (-0.15200309344504997f)   // log2(0.9)

// gfx12-family CPol: TH in bits [2:0]; 0 = RT, 1 = NT.
#define CPOL_RT 0
#define CPOL_NT 1

#if defined(__AMDGCN__) &&                                                 \
    __has_builtin(__builtin_amdgcn_global_load_async_to_lds_b128) &&       \
    __has_builtin(__builtin_amdgcn_global_store_async_from_lds_b128) &&    \
    __has_builtin(__builtin_amdgcn_s_wait_asynccnt)
#define USE_ASYNC_LDS 1
#else
#define USE_ASYNC_LDS 0
#endif

// ------------------- Pass 1a: pure stream copy (param, mom) ----------------
__global__ __launch_bounds__(BLOCK) void sgd_stream_copy(
    const float* __restrict__ src, float* __restrict__ dst, long long n) {
  const long long n4 = n >> 2;
  const long long gid = (long long)blockIdx.x * BLOCK + threadIdx.x;
  const long long gstride = (long long)gridDim.x * BLOCK;

#if USE_ASYNC_LDS
  // Async DMA: global -> LDS -> global, 16B/lane, 4 staging buffers (16 KB).
  // s_wait_asynccnt 0 at type boundaries (loads vs stores complete out of
  // order against the shared ASYNCcnt, so 0 is the only safe cross-type
  // fence); cross-wave interleaving provides the read/write overlap.
  __shared__ __align__(16) unsigned char lbuf[4][BLOCK * 16];
  const v4f* s4 = (const v4f*)src;
  v4f* d4 = (v4f*)dst;
  const long long super = 4LL * BLOCK;  // float4 units per block iteration
  for (long long s = (long long)blockIdx.x * super; s < n4;
       s += (long long)gridDim.x * super) {
#pragma unroll
    for (int b = 0; b < 4; ++b) {
      long long i = s + (long long)b * BLOCK + threadIdx.x;
      if (i < n4)
        __builtin_amdgcn_global_load_async_to_lds_b128(
            (g_v4i*)(s4 + i), (l_v4i*)&lbuf[b][threadIdx.x * 16], 0, CPOL_RT);
    }
    __builtin_amdgcn_s_wait_asynccnt(0);
#pragma unroll
    for (int b = 0; b < 4; ++b) {
      long long i = s + (long long)b * BLOCK + threadIdx.x;
      if (i < n4)
        __builtin_amdgcn_global_store_async_from_lds_b128(
            (g_v4i*)(d4 + i), (l_v4i*)&lbuf[b][threadIdx.x * 16], 0, CPOL_NT);
    }
    __builtin_amdgcn_s_wait_asynccnt(0);  // buffers reused next iteration
  }
#else
  // Fallback: b128 RT loads (keep inputs warm in L2 for the scatter pass),
  // non-temporal b128 stores (outputs are never re-read).
  const v4f* s4 = (const v4f*)src;
  v4f* d4 = (v4f*)dst;
  for (long long i = gid; i < n4; i += gstride) {
    v4f v = s4[i];
    __builtin_nontemporal_store(v, d4 + i);
  }
#endif
  // scalar tail (n % 4 != 0 safety)
  for (long long i = (n4 << 2) + gid; i < n; i += gstride) dst[i] = src[i];
}

// ----------------- Pass 1b: last_update int -> float conversion ------------
__global__ __launch_bounds__(BLOCK) void sgd_lu_convert(
    const int* __restrict__ lu, float* __restrict__ out_lu, long long n) {
  const long long n4 = n >> 2;
  const long long gid = (long long)blockIdx.x * BLOCK + threadIdx.x;
  const long long gstride = (long long)gridDim.x * BLOCK;
  const v4i* l4 = (const v4i*)lu;
  v4f* ol4 = (v4f*)out_lu;
  for (long long i = gid; i < n4; i += gstride) {
    v4i lv = l4[i];  // RT: lu is re-read by the scatter pass
    v4f lf = __builtin_convertvector(lv, v4f);
    __builtin_nontemporal_store(lf, ol4 + i);
  }
  for (long long i = (n4 << 2) + gid; i < n; i += gstride)
    out_lu[i] = (float)lu[i];
}

// ----------------------- Pass 2: gather/update/scatter ---------------------
__global__ __launch_bounds__(BLOCK) void sgd_scatter_update(
    const float* __restrict__ param, const float* __restrict__ gval,
    const int* __restrict__ gidx, const float* __restrict__ mom,
    const int* __restrict__ lu, const int* __restrict__ iter_ptr,
    float* __restrict__ out_param, float* __restrict__ out_mom,
    float* __restrict__ out_lu, int nnz) {
  const int i = blockIdx.x * BLOCK + threadIdx.x;
  if (i >= nnz) return;
  const int it = iter_ptr[0];  // uniform -> s_load
  // gidx/gval are streamed exactly once: NT loads, don't evict gather lines.
  const int idx = __builtin_nontemporal_load(gidx + i);
  const float gv = __builtin_nontemporal_load(gval + i);
  // Random gathers: RT so the copy pass's L2 residency pays off.
  const float p = param[idx];
  const float b = mom[idx];
  const int l = lu[idx];
  const float g = __builtin_fmaf(K_WD, p, gv);                 // weight decay
  const float mf = exp2f((float)(it - l) * K_LOG2_MOM);        // 0.9^(it-l)
  const float nb = __builtin_fmaf(mf, b, g);                   // buf update
  // Scattered write-once stores: NT (never re-read by this dispatch).
  __builtin_nontemporal_store(nb, out_mom + idx);
  __builtin_nontemporal_store(__builtin_fmaf(-K_LR, nb, p), out_param + idx);
  __builtin_nontemporal_store((float)it, out_lu + idx);
}

// ------------------------------- launcher ----------------------------------
extern "C" void kernel_launch(void* const* d_in, const int* in_sizes, int n_in,
                              void* d_out, int out_size, void* d_ws,
                              size_t ws_size, hipStream_t stream) {
  const float* param = (const float*)d_in[0];
  const float* gval  = (const float*)d_in[1];
  const int*   gidx  = (const int*)d_in[2];
  const float* mom   = (const float*)d_in[3];
  const int*   lu    = (const int*)d_in[4];
  const int*   iter  = (const int*)d_in[5];

  const long long P = in_sizes[0];
  const int NNZ = in_sizes[1];

  float* out_param = (float*)d_out;
  float* out_mom   = out_param + P;
  float* out_lu    = out_mom + P;

  const int copy_blocks = 4096;  // grid-stride; plenty of waves to fill HBM
  sgd_stream_copy<<<copy_blocks, BLOCK, 0, stream>>>(param, out_param, P);
  sgd_stream_copy<<<copy_blocks, BLOCK, 0, stream>>>(mom, out_mom, P);
  sgd_lu_convert<<<copy_blocks, BLOCK, 0, stream>>>(lu, out_lu, P);

  const int sc_blocks = (NNZ + BLOCK - 1) / BLOCK;
  sgd_scatter_update<<<sc_blocks, BLOCK, 0, stream>>>(
      param, gval, gidx, mom, lu, iter, out_param, out_mom, out_lu, NNZ);
}